// MuJoD_88751204204869
// MI455X (gfx1250) — compile-verified
//
#include <hip/hip_runtime.h>
#include <hip/hip_bf16.h>

#define NENT 15000
#define NREL 474
#define DD   256
#define KK   8
#define HH   2
#define BB   1024

typedef float v2f __attribute__((ext_vector_type(2)));
typedef float v8f __attribute__((ext_vector_type(8)));

// D(16x16,f32) += A(16x4,f32) x B(4x16,f32), fp32 accumulate (native fp32 tensor path)
__device__ __forceinline__ v8f wmma4(v2f a, v2f b, v8f c) {
  return __builtin_amdgcn_wmma_f32_16x16x4_f32(false, a, false, b, (short)0, c, false, false);
}

// Async bulk copy: 16 bytes global -> LDS, tracked by ASYNCcnt.
__device__ __forceinline__ void async_b128(uint32_t lds_off, const float* gaddr) {
  asm volatile("global_load_async_to_lds_b128 %0, %1, off" :: "v"(lds_off), "v"(gaddr) : "memory");
}
__device__ __forceinline__ uint32_t lds_off(const void* p) {
  return (uint32_t)(uintptr_t)p;   // flat LDS aperture: addr[31:0] = LDS byte address
}

// ---------------------------------------------------------------------------
// Expert gate weights: sc = x @ exps_T[m] (optionally / sigmoid(relgate)),
// softmax over K. One thread per (row, k).
// ---------------------------------------------------------------------------
__global__ void k_weights(const float* __restrict__ ent, const int* __restrict__ bd,
                          const float* __restrict__ expsT, const float* __restrict__ relgate,
                          float* __restrict__ wout, int nrows, int gathered)
{
  const int m  = blockIdx.y;
  const int t  = threadIdx.x;
  const int rl = t >> 3, k = t & 7;
  const int n  = blockIdx.x * 32 + rl;
  __shared__ float sc[32][KK];
  float s = 0.0f;
  if (n < nrows) {
    const float* xr;
    float gmul = 1.0f;
    if (gathered) {
      const int srow = bd[2 * n];
      const int rrow = bd[2 * n + 1];
      xr = ent + ((size_t)m * NENT + srow) * DD;
      gmul = 1.0f + __expf(-relgate[rrow]);   // 1/sigmoid(g)
    } else {
      xr = ent + ((size_t)m * NENT + n) * DD;
    }
    const float* Tm = expsT + (size_t)m * DD * KK;
    for (int d = 0; d < DD; ++d) s += xr[d] * Tm[d * KK + k];
    s *= gmul;
  }
  sc[rl][k] = s;
  __syncthreads();
  if (n < nrows) {
    float mx = sc[rl][0];
    #pragma unroll
    for (int j = 1; j < KK; ++j) mx = fmaxf(mx, sc[rl][j]);
    float sum = 0.0f;
    #pragma unroll
    for (int j = 0; j < KK; ++j) sum += __expf(sc[rl][j] - mx);
    wout[((size_t)m * nrows + n) * KK + k] = __expf(s - mx) / sum;
  }
}

// bc[m,k,o] = sum_d expb[m,k,d] * expW[m,k,o,d]
__global__ void k_bc(const float* __restrict__ expW, const float* __restrict__ expb,
                     float* __restrict__ bc)
{
  const int mk = blockIdx.x;
  const int o  = threadIdx.x;
  const float* W = expW + ((size_t)mk * DD + o) * DD;
  const float* b = expb + (size_t)mk * DD;
  float s = 0.0f;
  for (int d = 0; d < DD; ++d) s += b[d] * W[d];
  bc[(size_t)mk * DD + o] = s;
}

// ---------------------------------------------------------------------------
// Batch MoE: eo[n,k,o] = x[n]·W[m,k,o,:] - bc   (written to e_exp output)
//            e_mixed[n,o] = sum_k w[n,k]*eo[n,k,o]
// 16 waves per block. Gathered A rows staged into LDS via async-to-LDS B128.
// ---------------------------------------------------------------------------
__global__ void k_moe_batch(const float* __restrict__ ent, const int* __restrict__ bd,
                            const float* __restrict__ expW, const float* __restrict__ bc,
                            const float* __restrict__ wts,
                            float* __restrict__ e_exp, float* __restrict__ e_mixed)
{
  __shared__ float xs[16][DD];  // 16KB
  const int m  = blockIdx.y;
  const int m0 = blockIdx.x * 16;

  // async gather: 16 rows x 256 f32 = 1024 B128 chunks, 512 threads x 2
  {
    const uint32_t base = lds_off(&xs[0][0]);
    #pragma unroll
    for (int q = 0; q < 2; ++q) {
      const int cid = threadIdx.x + 512 * q;
      const int rr  = cid >> 6;
      const int cc  = (cid & 63) * 4;
      const int srow = bd[2 * (m0 + rr)];
      async_b128(base + (uint32_t)(rr * DD + cc) * 4u,
                 ent + ((size_t)m * NENT + srow) * DD + cc);
    }
    asm volatile("s_wait_asynccnt 0x0" ::: "memory");
    __syncthreads();
  }

  const int wave = threadIdx.x >> 5, lane = threadIdx.x & 31;
  const int half = lane >> 4, lr = lane & 15;
  const int o0   = wave * 16;
  const float* Wm = expW + (size_t)m * KK * DD * DD;
  v8f acc = {};
  for (int k = 0; k < KK; ++k) {
    const float* wk = Wm + (size_t)k * DD * DD + (size_t)(o0 + lr) * DD + 2 * half;
    v8f c = {};
    #pragma unroll 8
    for (int d0 = 0; d0 < DD; d0 += 4) {
      v2f a = *(const v2f*)(&xs[lr][d0 + 2 * half]);
      v2f b = *(const v2f*)(wk + d0);
      c = wmma4(a, b, c);
    }
    const float bck = bc[((size_t)m * KK + k) * DD + o0 + lr];
    #pragma unroll
    for (int v = 0; v < 8; ++v) {
      const int grow = m0 + v + 8 * half;
      const float eo = c[v] - bck;
      e_exp[(size_t)m * BB * KK * DD + ((size_t)grow * KK + k) * DD + o0 + lr] = eo;
      acc[v] += wts[((size_t)m * BB + grow) * KK + k] * eo;
    }
  }
  #pragma unroll
  for (int v = 0; v < 8; ++v)
    e_mixed[((size_t)m * BB + m0 + v + 8 * half) * DD + o0 + lr] = acc[v];
}

// All-entity MoE mix (no gate, no eo output), rows clamped at NENT.
__global__ void k_moe_all(const float* __restrict__ ent, const float* __restrict__ expW,
                          const float* __restrict__ bc, const float* __restrict__ wts,
                          float* __restrict__ all_mix)
{
  __shared__ float xs[16][DD];  // 16KB
  const int m  = blockIdx.y;
  const int n0 = blockIdx.x * 16;

  {
    const uint32_t base = lds_off(&xs[0][0]);
    #pragma unroll
    for (int q = 0; q < 2; ++q) {
      const int cid = threadIdx.x + 512 * q;
      const int rr  = cid >> 6;
      const int cc  = (cid & 63) * 4;
      const int row = min(n0 + rr, NENT - 1);
      async_b128(base + (uint32_t)(rr * DD + cc) * 4u,
                 ent + ((size_t)m * NENT + row) * DD + cc);
    }
    asm volatile("s_wait_asynccnt 0x0" ::: "memory");
    __syncthreads();
  }

  const int wave = threadIdx.x >> 5, lane = threadIdx.x & 31;
  const int half = lane >> 4, lr = lane & 15;
  const int o0   = wave * 16;
  const float* Wm = expW + (size_t)m * KK * DD * DD;
  v8f acc = {};
  for (int k = 0; k < KK; ++k) {
    const float* wk = Wm + (size_t)k * DD * DD + (size_t)(o0 + lr) * DD + 2 * half;
    v8f c = {};
    #pragma unroll 8
    for (int d0 = 0; d0 < DD; d0 += 4) {
      v2f a = *(const v2f*)(&xs[lr][d0 + 2 * half]);
      v2f b = *(const v2f*)(wk + d0);
      c = wmma4(a, b, c);
    }
    const float bck = bc[((size_t)m * KK + k) * DD + o0 + lr];
    #pragma unroll
    for (int v = 0; v < 8; ++v) {
      const int gc = min(n0 + v + 8 * half, NENT - 1);
      acc[v] += wts[((size_t)m * NENT + gc) * KK + k] * (c[v] - bck);
    }
  }
  #pragma unroll
  for (int v = 0; v < 8; ++v) {
    const int grow = n0 + v + 8 * half;
    if (grow < NENT)
      all_mix[((size_t)m * NENT + grow) * DD + o0 + lr] = acc[v];
  }
}

// ---------------------------------------------------------------------------
// Fusion: per head h, per modality m: Y = relu(es[m] @ W[m,h]^T + b[m,h]);
// sco = Y·att, softmax over m, out += sum_m wgt*Y ; summed over h. 48KB LDS.
// ---------------------------------------------------------------------------
__global__ void k_fusion(const float* __restrict__ src, const int* __restrict__ bd,
                         const float* __restrict__ W, const float* __restrict__ bias,
                         const float* __restrict__ att, float* __restrict__ out,
                         int nrows, int src_stride, int gathered)
{
  __shared__ float Ylds[3][16][DD];
  __shared__ float sco[16][3];
  __shared__ float wgt[16][3];
  const int n0   = blockIdx.x * 16;
  const int wave = threadIdx.x >> 5, lane = threadIdx.x & 31;
  const int half = lane >> 4, lr = lane & 15;
  const int o0   = wave * 16;
  const int rloc = min(n0 + lr, nrows - 1);
  const int srcrow = gathered ? bd[2 * rloc + 1] : rloc;

  float accout[8];
  #pragma unroll
  for (int j = 0; j < 8; ++j) accout[j] = 0.0f;

  for (int h = 0; h < HH; ++h) {
    for (int mm = 0; mm < 3; ++mm) {
      const float* xr  = src + ((size_t)mm * src_stride + srcrow) * DD + 2 * half;
      const float* Wmh = W + (((size_t)mm * HH + h) * DD + (o0 + lr)) * DD + 2 * half;
      v8f c = {};
      #pragma unroll 8
      for (int d0 = 0; d0 < DD; d0 += 4) {
        v2f a = *(const v2f*)(xr + d0);
        v2f b = *(const v2f*)(Wmh + d0);
        c = wmma4(a, b, c);
      }
      const float bv = bias[((size_t)mm * HH + h) * DD + o0 + lr];
      #pragma unroll
      for (int v = 0; v < 8; ++v) {
        float y = c[v] + bv;
        Ylds[mm][v + 8 * half][o0 + lr] = (y > 0.0f) ? y : 0.0f;
      }
    }
    __syncthreads();
    if (threadIdx.x < 48) {
      const int rr = threadIdx.x / 3, mm = threadIdx.x % 3;
      float s = 0.0f;
      for (int o = 0; o < DD; ++o) s += Ylds[mm][rr][o] * att[o];
      sco[rr][mm] = s;
    }
    __syncthreads();
    if (threadIdx.x < 16) {
      const int rr = threadIdx.x;
      const float a0 = sco[rr][0], a1 = sco[rr][1], a2 = sco[rr][2];
      const float mx = fmaxf(a0, fmaxf(a1, a2));
      const float e0 = __expf(a0 - mx), e1 = __expf(a1 - mx), e2 = __expf(a2 - mx);
      const float inv = 1.0f / (e0 + e1 + e2);
      wgt[rr][0] = e0 * inv; wgt[rr][1] = e1 * inv; wgt[rr][2] = e2 * inv;
    }
    __syncthreads();
    {
      const int rr = threadIdx.x >> 5;
      const int ob = (threadIdx.x & 31) * 8;
      #pragma unroll
      for (int j = 0; j < 8; ++j) {
        const int o = ob + j;
        accout[j] += wgt[rr][0] * Ylds[0][rr][o]
                   + wgt[rr][1] * Ylds[1][rr][o]
                   + wgt[rr][2] * Ylds[2][rr][o];
      }
    }
    __syncthreads();
  }
  {
    const int rr = threadIdx.x >> 5;
    const int ob = (threadIdx.x & 31) * 8;
    const int grow = n0 + rr;
    if (grow < nrows) {
      #pragma unroll
      for (int j = 0; j < 8; ++j) out[(size_t)grow * DD + ob + j] = accout[j];
    }
  }
}

// ---------------------------------------------------------------------------
// Predtail: o[b,j] = sum_{r,i} (r[b,r]*eBN[b,i]) * rT[mod,r,i,j], then BN-out.
// M=1024, K=65536, N=128-per-block GEMM. rT tiles (32 i x 128 j = 16KB) are
// streamed through a double-buffered LDS ring with async-to-LDS B128 copies,
// overlapped with WMMA consumption; ASYNCcnt in-order completion gives the
// s_wait_asynccnt 0x4 retirement guarantee. LDS: 16K e + 16K r + 32K ring.
// ---------------------------------------------------------------------------
__global__ void k_predtail(const float* __restrict__ e_mixed, const float* __restrict__ j_e,
                           const float* __restrict__ rel_tab, const int* __restrict__ bd,
                           const float* __restrict__ j_r, const float* __restrict__ rT,
                           const float* __restrict__ gi, const float* __restrict__ bi,
                           const float* __restrict__ go, const float* __restrict__ bo,
                           float* __restrict__ tails)
{
  __shared__ float elds[16][DD];        // 16KB
  __shared__ float rlds[16][DD];        // 16KB
  __shared__ float rtbuf[2][32][128];   // 32KB double-buffered rT tile ring
  const int mod   = blockIdx.y;
  const int m0    = blockIdx.x * 16;
  const int jbase = blockIdx.z * 128;
  const float bnf = 1.0f / sqrtf(1.0f + 1e-5f);

  for (int idx = threadIdx.x; idx < 16 * DD; idx += blockDim.x) {
    const int rl = idx >> 8, col = idx & 255;
    const int grow = m0 + rl;
    float ev, rv;
    if (mod < 3) {
      ev = e_mixed[((size_t)mod * BB + grow) * DD + col];
      const int rid = bd[2 * grow + 1];
      rv = rel_tab[((size_t)mod * NREL + rid) * DD + col];
    } else {
      ev = j_e[(size_t)grow * DD + col];
      rv = j_r[(size_t)grow * DD + col];
    }
    elds[rl][col] = ev * (gi[(size_t)mod * DD + col] * bnf) + bi[(size_t)mod * DD + col];
    rlds[rl][col] = rv;
  }

  const int wave = threadIdx.x >> 5, lane = threadIdx.x & 31;
  const int half = lane >> 4, lr = lane & 15;
  const int jloc = wave * 16 + lr;                 // 0..127 within slice
  const float* rtm = rT + (size_t)mod * DD * DD * DD + jbase;
  const uint32_t rb0 = lds_off(&rtbuf[0][0][0]);
  const uint32_t rb1 = lds_off(&rtbuf[1][0][0]);

  // fill block blk (r = blk>>3, ib = blk&7): rT[r, ib*32 .. +32, jbase..+128]
  auto issue_fill = [&](int blk, uint32_t base) {
    const int r = blk >> 3, ib = blk & 7;
    const float* src = rtm + (size_t)r * DD * DD + (size_t)(ib * 32) * DD;
    #pragma unroll
    for (int q = 0; q < 4; ++q) {                  // 1024 B128 chunks / 256 thr
      const int cid = threadIdx.x + 256 * q;
      const int ii  = cid >> 5;                    // 0..31
      const int jj  = (cid & 31) * 4;              // 0..124
      async_b128(base + (uint32_t)(ii * 128 + jj) * 4u, src + (size_t)ii * DD + jj);
    }
  };

  issue_fill(0, rb0);
  __syncthreads();   // e/r staging visible to all

  v8f c = {};
  for (int blk = 0; blk < 2048; ++blk) {
    const uint32_t cur = (blk & 1) ? rb1 : rb0;
    if (blk < 2047) {
      issue_fill(blk + 1, (blk & 1) ? rb0 : rb1);
      asm volatile("s_wait_asynccnt 0x4" ::: "memory");  // fill(blk) retired
    } else {
      asm volatile("s_wait_asynccnt 0x0" ::: "memory");
    }
    __syncthreads();                               // all waves' fill(blk) done
    const int r = blk >> 3, ib = blk & 7;
    const float rv = rlds[lr][r];
    const float* buf = (blk & 1) ? &rtbuf[1][0][0] : &rtbuf[0][0][0];
    (void)cur;
    #pragma unroll
    for (int s = 0; s < 8; ++s) {
      const int il = s * 4 + 2 * half;
      const int ig = ib * 32 + il;
      v2f a; a.x = rv * elds[lr][ig]; a.y = rv * elds[lr][ig + 1];
      v2f b; b.x = buf[il * 128 + jloc]; b.y = buf[(il + 1) * 128 + jloc];
      c = wmma4(a, b, c);
    }
    __syncthreads();                               // consumption done; ring reusable
  }

  #pragma unroll
  for (int v = 0; v < 8; ++v) {
    const int grow = m0 + v + 8 * half;
    const int o = jbase + jloc;
    tails[((size_t)mod * BB + grow) * DD + o] =
        c[v] * (go[(size_t)mod * DD + o] * bnf) + bo[(size_t)mod * DD + o];
  }
}

// ---------------------------------------------------------------------------
// Scores: sigmoid(tails[mod] @ Z^T), Z in {e_all, all_mix[1], all_mix[2], e_all}
// ---------------------------------------------------------------------------
__global__ void k_scores(const float* __restrict__ tails, const float* __restrict__ e_all,
                         const float* __restrict__ all_mix, float* __restrict__ scores)
{
  const int mod  = blockIdx.z;
  const int m0   = blockIdx.y * 16;
  const int wave = threadIdx.x >> 5, lane = threadIdx.x & 31;
  const int half = lane >> 4, lr = lane & 15;
  const int ntile = blockIdx.x * 8 + wave;
  if (ntile * 16 >= NENT) return;   // wave-uniform
  const int n0 = ntile * 16;
  const float* Z = (mod == 1) ? all_mix + (size_t)1 * NENT * DD
                 : (mod == 2) ? all_mix + (size_t)2 * NENT * DD
                              : e_all;
  const float* ta = tails + ((size_t)mod * BB + m0 + lr) * DD + 2 * half;
  const int zr = min(n0 + lr, NENT - 1);
  const float* zp = Z + (size_t)zr * DD + 2 * half;
  v8f c = {};
  #pragma unroll 8
  for (int d0 = 0; d0 < DD; d0 += 4) {
    v2f a = *(const v2f*)(ta + d0);
    v2f b = *(const v2f*)(zp + d0);
    c = wmma4(a, b, c);
  }
  #pragma unroll
  for (int v = 0; v < 8; ++v) {
    const int grow = m0 + v + 8 * half;
    const int col = n0 + lr;
    if (col < NENT)
      scores[((size_t)mod * BB + grow) * NENT + col] = 1.0f / (1.0f + __expf(-c[v]));
  }
}

// ---------------------------------------------------------------------------
extern "C" void kernel_launch(void* const* d_in, const int* in_sizes, int n_in,
                              void* d_out, int out_size, void* d_ws, size_t ws_size,
                              hipStream_t stream)
{
  const int*   bd      = (const int*)  d_in[0];
  const float* ent_tab = (const float*)d_in[1];
  const float* rel_tab = (const float*)d_in[2];
  const float* relgate = (const float*)d_in[3];
  const float* exps_T  = (const float*)d_in[4];
  const float* expW    = (const float*)d_in[5];
  const float* expb    = (const float*)d_in[6];
  const float* ef_W    = (const float*)d_in[7];
  const float* ef_b    = (const float*)d_in[8];
  const float* ef_att  = (const float*)d_in[9];
  const float* rf_W    = (const float*)d_in[10];
  const float* rf_b    = (const float*)d_in[11];
  const float* rf_att  = (const float*)d_in[12];
  const float* rT      = (const float*)d_in[13];
  const float* bn_gin  = (const float*)d_in[14];
  const float* bn_bin  = (const float*)d_in[15];
  const float* bn_gout = (const float*)d_in[16];
  const float* bn_bout = (const float*)d_in[17];

  float* out    = (float*)d_out;
  float* scores = out;                              // (4, B, NENT)
  float* e_exp  = out + (size_t)4 * BB * NENT;      // (3, B, K, D)

  float* w = (float*)d_ws;
  size_t off = 0;
  float* wb    = w + off; off += (size_t)3 * BB * KK;
  float* wa    = w + off; off += (size_t)3 * NENT * KK;
  float* bc    = w + off; off += (size_t)3 * KK * DD;
  float* emix  = w + off; off += (size_t)3 * BB * DD;
  float* amix  = w + off; off += (size_t)3 * NENT * DD;
  float* je    = w + off; off += (size_t)BB * DD;
  float* jr    = w + off; off += (size_t)BB * DD;
  float* eall  = w + off; off += (size_t)NENT * DD;
  float* tails = w + off; off += (size_t)4 * BB * DD;
  (void)off; (void)ws_size; (void)in_sizes; (void)n_in; (void)out_size;

  k_bc<<<dim3(3 * KK), 256, 0, stream>>>(expW, expb, bc);
  k_weights<<<dim3(BB / 32, 3), 256, 0, stream>>>(ent_tab, bd, exps_T, relgate, wb, BB, 1);
  k_weights<<<dim3((NENT + 31) / 32, 3), 256, 0, stream>>>(ent_tab, bd, exps_T, relgate, wa, NENT, 0);
  k_moe_batch<<<dim3(BB / 16, 3), 512, 0, stream>>>(ent_tab, bd, expW, bc, wb, e_exp, emix);
  k_moe_all<<<dim3((NENT + 15) / 16, 3), 512, 0, stream>>>(ent_tab, expW, bc, wa, amix);
  k_fusion<<<dim3(BB / 16), 512, 0, stream>>>(emix, bd, ef_W, ef_b, ef_att, je, BB, BB, 0);
  k_fusion<<<dim3(BB / 16), 512, 0, stream>>>(rel_tab, bd, rf_W, rf_b, rf_att, jr, BB, NREL, 1);
  k_fusion<<<dim3((NENT + 15) / 16), 512, 0, stream>>>(amix, bd, ef_W, ef_b, ef_att, eall, NENT, NENT, 0);
  k_predtail<<<dim3(BB / 16, 4, 2), 256, 0, stream>>>(emix, je, rel_tab, bd, jr, rT,
                                                      bn_gin, bn_bin, bn_gout, bn_bout, tails);
  k_scores<<<dim3(118, BB / 16, 4), 256, 0, stream>>>(tails, eall, amix, scores);
}